// DOTGATConvDGL_75393855913985
// MI455X (gfx1250) — compile-verified
//
#include <hip/hip_runtime.h>
#include <math.h>

// ---------------------------------------------------------------------------
// DotGAT (DGL DotGatConv) for MI455X / gfx1250.
//   h = feat @ W^T          -> f32 WMMA GEMM (V_WMMA_F32_16X16X4_F32)
//   e = <h[src],h[dst]>/√D  -> edge pass 1 (+ segment max via float atomics)
//   softmax over dst        -> edge pass 2 (exp + segment sum)
//   out = Σ sa * h[src]     -> edge pass 3 (scatter atomicAdd, L2-resident)
// h (102.4 MB) fits in the 192 MB L2, so all random edge gathers hit L2.
// ---------------------------------------------------------------------------

typedef __attribute__((ext_vector_type(2))) float v2f;
typedef __attribute__((ext_vector_type(8))) float v8f;

#define N_NODES   100000
#define N_EDGES   1600000
#define IN_SIZE   256
#define NUM_HEADS 8
#define HEAD_DIM  32
#define OUT_COLS  256            // NUM_HEADS * HEAD_DIM

#define LDS_STRIDE 260           // 256 + 4 pad -> lanes hit distinct LDS banks
#define LDS_FLOATS (256 * LDS_STRIDE)
#define LDS_BYTES  (LDS_FLOATS * 4)   // 266,240 B <= 320 KB WGP LDS

// ---------------------------------------------------------------------------
// GEMM: h[N,256] = feat[N,256] @ W^T,  W is [256 out, 256 in] row-major.
// Block = 8 waves; whole W staged in LDS once per block (CDNA5 320KB LDS).
// Each wave computes a 16-row x 256-col output slab with 16 f32 WMMA tiles.
// ---------------------------------------------------------------------------
__global__ void __launch_bounds__(256) gemm_wmma_f32(
    const float* __restrict__ feat, const float* __restrict__ W,
    float* __restrict__ h) {
  extern __shared__ float lw[];
  const int tid = threadIdx.x;

  // Cooperative load of W into padded LDS (row stride 260 floats).
  const float4* Wv = (const float4*)W;
  for (int i4 = tid; i4 < (256 * 256) / 4; i4 += 256) {
    const int o = i4 >> 6;        // output row
    const int c = i4 & 63;        // float4 column
    *(float4*)(lw + o * LDS_STRIDE + (c << 2)) = Wv[i4];
  }
  __syncthreads();

  const int wave = tid >> 5;
  const int lane = tid & 31;
  const int tile = blockIdx.x * 8 + wave;         // 16-row tile index
  if (tile >= N_NODES / 16) return;

  const int m0   = tile * 16;
  const int mrow = lane & 15;                     // M (for A) / N (for B)
  const int koff = (lane >> 4) << 1;              // K sub-offset: 0 or 2
  const float* arow = feat + (size_t)(m0 + mrow) * IN_SIZE + koff;

  v8f acc[16] = {};                               // 16 N-tiles of 16x16 f32

  for (int k = 0; k < IN_SIZE; k += 4) {
    // A frag 16x4: lane(0-15)=rows M, K={koff,koff+1}
    v2f a = *(const v2f*)(arow + k);
#pragma unroll
    for (int n = 0; n < 16; ++n) {
      // B frag 4x16: B[k][n] = W[n][k]; lane gives N, VGPRs give K pair
      v2f b = *(const v2f*)(lw + (n * 16 + mrow) * LDS_STRIDE + k + koff);
      acc[n] = __builtin_amdgcn_wmma_f32_16x16x4_f32(
          /*neg_a=*/false, a, /*neg_b=*/false, b,
          /*c_mod=*/(short)0, acc[n], /*reuse_a=*/false, /*reuse_b=*/false);
    }
  }

  // D layout: lane L, vgpr v -> row m0 + v + 8*(L>=16), col n*16 + (L&15)
  const int rhi = (lane >> 4) << 3;
#pragma unroll
  for (int n = 0; n < 16; ++n) {
#pragma unroll
    for (int v = 0; v < 8; ++v) {
      h[(size_t)(m0 + rhi + v) * OUT_COLS + n * 16 + mrow] = acc[n][v];
    }
  }
}

// ---------------------------------------------------------------------------
// Init: out = 0, m = -inf, s = 0 (d_out / d_ws are poisoned by harness).
// ---------------------------------------------------------------------------
__global__ void init_kernel(float* __restrict__ out, float* __restrict__ m,
                            float* __restrict__ s) {
  const int i = blockIdx.x * blockDim.x + threadIdx.x;
  if (i < N_NODES * OUT_COLS) out[i] = 0.0f;
  if (i < N_NODES * NUM_HEADS) {
    m[i] = -INFINITY;
    s[i] = 0.0f;
  }
}

// Monotone float atomic max: positives compare as ints, negatives as
// reversed uints; -inf init (0xFF800000) is the max uint / min int. OK both ways.
__device__ __forceinline__ void atomic_max_f32(float* addr, float val) {
  if (val >= 0.0f)
    atomicMax((int*)addr, __float_as_int(val));
  else
    atomicMin((unsigned int*)addr, __float_as_uint(val));
}

// ---------------------------------------------------------------------------
// Pass 1: per-(edge,head) dot score + segment max over dst.
// ---------------------------------------------------------------------------
__global__ void score_kernel(const float* __restrict__ h,
                             const int* __restrict__ src,
                             const int* __restrict__ dst,
                             float* __restrict__ e, float* __restrict__ m) {
  const int t = blockIdx.x * blockDim.x + threadIdx.x;
  if (t >= N_EDGES * NUM_HEADS) return;
  const int edge = t >> 3, hd = t & 7;
  const int si = src[edge], di = dst[edge];
  const float4* a = (const float4*)(h + (size_t)si * OUT_COLS + hd * HEAD_DIM);
  const float4* b = (const float4*)(h + (size_t)di * OUT_COLS + hd * HEAD_DIM);
  float sum = 0.0f;
#pragma unroll
  for (int j = 0; j < HEAD_DIM / 4; ++j) {
    const float4 x = a[j], y = b[j];
    sum += x.x * y.x + x.y * y.y + x.z * y.z + x.w * y.w;
  }
  sum *= 0.17677669529663689f;  // 1/sqrt(32)
  e[t] = sum;
  atomic_max_f32(&m[di * NUM_HEADS + hd], sum);
}

// ---------------------------------------------------------------------------
// Pass 2: ex = exp(e - m[dst]); segment sum over dst. e is overwritten w/ ex.
// ---------------------------------------------------------------------------
__global__ void expsum_kernel(const int* __restrict__ dst,
                              float* __restrict__ e,
                              const float* __restrict__ m,
                              float* __restrict__ s) {
  const int t = blockIdx.x * blockDim.x + threadIdx.x;
  if (t >= N_EDGES * NUM_HEADS) return;
  const int edge = t >> 3, hd = t & 7;
  const int di = dst[edge];
  const float ex = __expf(e[t] - m[di * NUM_HEADS + hd]);
  e[t] = ex;
  atomicAdd(&s[di * NUM_HEADS + hd], ex);
}

// ---------------------------------------------------------------------------
// Pass 3: out[dst,hd,:] += (ex / s[dst,hd]) * h[src,hd,:]  (32 f32 atomics).
// ---------------------------------------------------------------------------
__global__ void agg_kernel(const float* __restrict__ h,
                           const int* __restrict__ src,
                           const int* __restrict__ dst,
                           const float* __restrict__ e,
                           const float* __restrict__ s,
                           float* __restrict__ out) {
  const int t = blockIdx.x * blockDim.x + threadIdx.x;
  if (t >= N_EDGES * NUM_HEADS) return;
  const int edge = t >> 3, hd = t & 7;
  const int si = src[edge], di = dst[edge];
  const float w = e[t] / s[di * NUM_HEADS + hd];
  const float* hv = h + (size_t)si * OUT_COLS + hd * HEAD_DIM;
  float* op = out + (size_t)di * OUT_COLS + hd * HEAD_DIM;
#pragma unroll
  for (int d = 0; d < HEAD_DIM; ++d) atomicAdd(&op[d], hv[d] * w);
}

// ---------------------------------------------------------------------------
extern "C" void kernel_launch(void* const* d_in, const int* in_sizes, int n_in,
                              void* d_out, int out_size, void* d_ws,
                              size_t ws_size, hipStream_t stream) {
  const float* feat = (const float*)d_in[0];
  const float* W    = (const float*)d_in[1];
  const int*   src  = (const int*)d_in[2];
  const int*   dst  = (const int*)d_in[3];
  float* out = (float*)d_out;

  // Workspace layout (bytes):
  //   h : [0, 102,400,000)            N*256 f32
  //   e : [102.4e6, 153.6e6)          E*8 f32 (scores, then exp)
  //   m : [153.6e6, 156.8e6)          N*8 f32 segment max
  //   s : [156.8e6, 160.0e6)          N*8 f32 segment sum
  char* ws = (char*)d_ws;
  float* h = (float*)ws;
  float* e = (float*)(ws + (size_t)N_NODES * OUT_COLS * sizeof(float));
  float* m = (float*)(ws + (size_t)N_NODES * OUT_COLS * sizeof(float)
                         + (size_t)N_EDGES * NUM_HEADS * sizeof(float));
  float* s = m + (size_t)N_NODES * NUM_HEADS;

  // Allow 260 KB of dynamic LDS (CDNA5 WGP has 320 KB).
  hipFuncSetAttribute((const void*)gemm_wmma_f32,
                      hipFuncAttributeMaxDynamicSharedMemorySize, LDS_BYTES);

  const int ntiles = N_NODES / 16;                     // 6250
  gemm_wmma_f32<<<(ntiles + 7) / 8, 256, LDS_BYTES, stream>>>(feat, W, h);

  init_kernel<<<(N_NODES * OUT_COLS + 255) / 256, 256, 0, stream>>>(out, m, s);

  const int EH = N_EDGES * NUM_HEADS;                  // 12.8M
  score_kernel <<<(EH + 255) / 256, 256, 0, stream>>>(h, src, dst, e, m);
  expsum_kernel<<<(EH + 255) / 256, 256, 0, stream>>>(dst, e, m, s);
  agg_kernel   <<<(EH + 255) / 256, 256, 0, stream>>>(h, src, dst, e, s, out);
}